// GHMR_429496730148
// MI455X (gfx1250) — compile-verified
//
#include <hip/hip_runtime.h>

#define TPB   256
#define NBLK  4096
#define BINS  30
#define MU    0.02f

typedef float v2f __attribute__((ext_vector_type(2)));
typedef float v4f __attribute__((ext_vector_type(4)));
typedef float v8f __attribute__((ext_vector_type(8)));

// ---------------------------------------------------------------------------
// Wave-wide sum of 8 floats/lane (256 values) using the matrix pipe.
// Treat the per-lane v8f as four 16x4 f32 A-matrices (2 VGPRs each) and
// multiply by an all-ones 4x16 B: D[m][n] = sum_k A[m][k], accumulated over
// the four chunks. Each register slot maps to a unique (m,k), so every value
// is counted exactly once. D's row-sums are replicated across columns, so
// d[0..7] in lanes 0-15 hold s[0..7] and lanes 16-31 hold s[8..15]; one
// cross-half shuffle finishes the reduction. Requires EXEC == all 1s.
// ---------------------------------------------------------------------------
__device__ __forceinline__ float wave_sum_wmma(v8f x) {
    const v2f ones = {1.0f, 1.0f};
    v8f d = {0.f, 0.f, 0.f, 0.f, 0.f, 0.f, 0.f, 0.f};
    v2f a0; a0[0] = x[0]; a0[1] = x[1];
    v2f a1; a1[0] = x[2]; a1[1] = x[3];
    v2f a2; a2[0] = x[4]; a2[1] = x[5];
    v2f a3; a3[0] = x[6]; a3[1] = x[7];
    d = __builtin_amdgcn_wmma_f32_16x16x4_f32(false, a0, false, ones, (short)0, d, false, false);
    d = __builtin_amdgcn_wmma_f32_16x16x4_f32(false, a1, false, ones, (short)0, d, false, false);
    d = __builtin_amdgcn_wmma_f32_16x16x4_f32(false, a2, false, ones, (short)0, d, false, false);
    d = __builtin_amdgcn_wmma_f32_16x16x4_f32(false, a3, false, ones, (short)0, d, false, false);
    float t = ((d[0] + d[1]) + (d[2] + d[3])) + ((d[4] + d[5]) + (d[6] + d[7]));
    t += __shfl_xor(t, 16, 32);   // combine s[0..7] half with s[8..15] half
    return t;                     // full 256-value sum, broadcast to all lanes
}

__device__ __forceinline__ int bin_of(float diff) {
    float g = fabsf(diff) * rsqrtf(diff * diff + MU * MU);  // in [0,1)
    int b = (int)(g * (float)BINS);                          // trunc toward 0
    return (b > BINS - 1) ? (BINS - 1) : b;
}

// ---------------------------------------------------------------------------
// Workspace layout (dword slots in d_ws):
//   [0..29]  counts (u32, atomic)     [30] sum_lw (f32, atomic)
//   [31]     loss_acc (f32, atomic)   [32..61] coef (f32)   [62] inv_tot
// ---------------------------------------------------------------------------

__global__ void ghmr_init(unsigned* ws) {
    if (threadIdx.x < 32) ws[threadIdx.x] = 0u;   // counts + sum_lw + loss_acc
}

// Pass 1: histogram of valid counts per gradient bin + sum(label_weight).
__global__ void ghmr_pass1(const float* __restrict__ pred,
                           const float* __restrict__ targ,
                           const float* __restrict__ lw,
                           unsigned* __restrict__ counts,
                           float* __restrict__ sum_lw,
                           long long n4, int ntail) {
    __shared__ unsigned hist[8 * 32];             // per-wave padded histograms
    const int tid  = threadIdx.x;
    const int wave = tid >> 5;
    for (int i = tid; i < 8 * 32; i += TPB) hist[i] = 0u;
    __syncthreads();

    const v4f* p4 = (const v4f*)pred;
    const v4f* t4 = (const v4f*)targ;
    const v4f* w4 = (const v4f*)lw;

    v8f acc = {0.f, 0.f, 0.f, 0.f, 0.f, 0.f, 0.f, 0.f};
    const long long stride = (long long)gridDim.x * TPB;
    for (long long i = (long long)blockIdx.x * TPB + tid; i < n4; i += stride) {
        v4f p = p4[i], t = t4[i], w = w4[i];
#pragma unroll
        for (int j = 0; j < 4; ++j) {
            float d = p[j] - t[j];
            int   b = bin_of(d);
            if (w[j] > 0.0f) atomicAdd(&hist[wave * 32 + b], 1u);
            acc[j] += w[j];
        }
    }
    if (blockIdx.x == 0 && tid < ntail) {         // scalar tail (N % 4)
        long long e = n4 * 4 + tid;
        float d = pred[e] - targ[e];
        float w = lw[e];
        int   b = bin_of(d);
        if (w > 0.0f) atomicAdd(&hist[wave * 32 + b], 1u);
        acc[0] += w;
    }
    __syncthreads();

    if (tid < BINS) {
        unsigned c = 0;
#pragma unroll
        for (int w = 0; w < 8; ++w) c += hist[w * 32 + tid];
        if (c) atomicAdd(&counts[tid], c);
    }
    float wsum = wave_sum_wmma(acc);              // uniform: all lanes reach this
    if ((tid & 31) == 0) atomicAdd(sum_lw, wsum);
}

// Mid: coef[b] = tot / max(counts[b],1) / max(n,1);  inv_tot = 1/tot.
__global__ void ghmr_mid(const unsigned* __restrict__ counts,
                         const float* __restrict__ sum_lw,
                         float* __restrict__ coef,
                         float* __restrict__ inv_tot) {
    int t = threadIdx.x;                          // one wave32
    unsigned c = (t < BINS) ? counts[t] : 0u;
    unsigned long long mask = __ballot(t < BINS && c > 0u);
    int   n   = __popcll(mask);
    float nn  = (float)((n > 1) ? n : 1);
    float tot = fmaxf(*sum_lw, 1.0f);
    if (t < BINS) coef[t] = tot / fmaxf((float)c, 1.0f) / nn;
    if (t == 0)   *inv_tot = 1.0f / tot;
}

// Pass 2: sum(loss * coef[idx] * valid). Non-temporal (last-use) loads.
__global__ void ghmr_pass2(const float* __restrict__ pred,
                           const float* __restrict__ targ,
                           const float* __restrict__ lw,
                           const float* __restrict__ coef,
                           float* __restrict__ loss_acc,
                           long long n4, int ntail) {
    __shared__ float scoef[32];
    const int tid = threadIdx.x;
    if (tid < 32) scoef[tid] = (tid < BINS) ? coef[tid] : 0.0f;
    __syncthreads();

    const v4f* p4 = (const v4f*)pred;
    const v4f* t4 = (const v4f*)targ;
    const v4f* w4 = (const v4f*)lw;

    v8f acc = {0.f, 0.f, 0.f, 0.f, 0.f, 0.f, 0.f, 0.f};
    const long long stride = (long long)gridDim.x * TPB;
    for (long long i = (long long)blockIdx.x * TPB + tid; i < n4; i += stride) {
        v4f p = __builtin_nontemporal_load(p4 + i);
        v4f t = __builtin_nontemporal_load(t4 + i);
        v4f w = __builtin_nontemporal_load(w4 + i);
#pragma unroll
        for (int j = 0; j < 4; ++j) {
            float d    = p[j] - t[j];
            float loss = sqrtf(d * d + MU * MU) - MU;
            int   b    = bin_of(d);
            acc[j] += (w[j] > 0.0f) ? loss * scoef[b] : 0.0f;
        }
    }
    if (blockIdx.x == 0 && tid < ntail) {
        long long e = n4 * 4 + tid;
        float d    = pred[e] - targ[e];
        float loss = sqrtf(d * d + MU * MU) - MU;
        int   b    = bin_of(d);
        acc[0] += (lw[e] > 0.0f) ? loss * scoef[b] : 0.0f;
    }

    float wsum = wave_sum_wmma(acc);
    if ((tid & 31) == 0) atomicAdd(loss_acc, wsum);
}

__global__ void ghmr_final(const float* __restrict__ loss_acc,
                           const float* __restrict__ inv_tot,
                           float* __restrict__ out) {
    out[0] = loss_acc[0] * inv_tot[0];            // LOSS_WEIGHT == REWEIGHT == 1
}

extern "C" void kernel_launch(void* const* d_in, const int* in_sizes, int n_in,
                              void* d_out, int out_size, void* d_ws, size_t ws_size,
                              hipStream_t stream) {
    const float* pred = (const float*)d_in[0];
    const float* targ = (const float*)d_in[1];
    const float* lw   = (const float*)d_in[2];

    long long N     = (long long)in_sizes[0];
    long long n4    = N >> 2;
    int       ntail = (int)(N & 3);

    unsigned* counts   = (unsigned*)d_ws;
    float*    fws      = (float*)d_ws;
    float*    sum_lw   = fws + 30;
    float*    loss_acc = fws + 31;
    float*    coef     = fws + 32;
    float*    inv_tot  = fws + 62;
    float*    out      = (float*)d_out;

    long long want   = (n4 + TPB - 1) / TPB;
    int       blocks = (want < (long long)NBLK) ? (int)(want ? want : 1) : NBLK;

    ghmr_init <<<1, 64, 0, stream>>>((unsigned*)d_ws);
    ghmr_pass1<<<blocks, TPB, 0, stream>>>(pred, targ, lw, counts, sum_lw, n4, ntail);
    ghmr_mid  <<<1, 32, 0, stream>>>(counts, sum_lw, coef, inv_tot);
    ghmr_pass2<<<blocks, TPB, 0, stream>>>(pred, targ, lw, coef, loss_acc, n4, ntail);
    ghmr_final<<<1, 1, 0, stream>>>(loss_acc, inv_tot, out);
}